// Model_16045997818200
// MI455X (gfx1250) — compile-verified
//
#include <hip/hip_runtime.h>
#include <hip/hip_bf16.h>

#define NNODES 100000
#define NEDGES 1600000
#define NGRAPHS 128
#define INF 128
#define HF 128
#define OUTF 64
#define HID 10
#define NCLS 10

typedef __attribute__((ext_vector_type(2))) float v2f;
typedef __attribute__((ext_vector_type(8))) float v8f;

__device__ __forceinline__ void atomAddF(float* p, float v) {
  __hip_atomic_fetch_add(p, v, __ATOMIC_RELAXED, __HIP_MEMORY_SCOPE_AGENT);
}

// ---------------- zero fill (float4 granularity) ----------------
__global__ void k_zero4(float4* p, long n4) {
  long i = (long)blockIdx.x * blockDim.x + threadIdx.x;
  if (i < n4) p[i] = make_float4(0.f, 0.f, 0.f, 0.f);
}

// ---------------- degree counting ----------------
__global__ void k_deg(const int* __restrict__ src, const int* __restrict__ dst,
                      int* degO, int* degI, int E) {
  int e = blockIdx.x * blockDim.x + threadIdx.x;
  if (e < E) {
    atomicAdd(&degO[src[e]], 1);
    atomicAdd(&degI[dst[e]], 1);
  }
}

__global__ void k_rsq(const int* __restrict__ degO, const int* __restrict__ degI,
                      float* rsO, float* rsI, int Nn) {
  int n = blockIdx.x * blockDim.x + threadIdx.x;
  if (n < Nn) {
    rsO[n] = rsqrtf((float)max(degO[n], 1));
    rsI[n] = rsqrtf((float)max(degI[n], 1));
  }
}

// ---------------- x = feats * rs_out (row scale), float4 ----------------
__global__ void k_scale(const float* __restrict__ in, const float* __restrict__ rs,
                        float* __restrict__ out, int Nn) {
  int i4 = blockIdx.x * blockDim.x + threadIdx.x;     // index into float4s, 32 per row
  if (i4 >= Nn * (INF / 4)) return;
  int node = i4 >> 5;
  float s = rs[node];
  float4 v = ((const float4*)in)[i4];
  v.x *= s; v.y *= s; v.z *= s; v.w *= s;
  ((float4*)out)[i4] = v;
}

// ---------------- edge gather + scatter-add: one wave per edge, 128 floats ----------------
__global__ void __launch_bounds__(256)
k_edge_agg(const float* __restrict__ x, const int* __restrict__ src,
           const int* __restrict__ dst, float* __restrict__ agg, int E) {
  int wave = (int)(((long)blockIdx.x * blockDim.x + threadIdx.x) >> 5);
  int lane = threadIdx.x & 31;
  if (wave >= E) return;
  int s = src[wave];
  int d = dst[wave];
  const float4 v = *(const float4*)&x[(size_t)s * HF + lane * 4];
  float* o = &agg[(size_t)d * HF + lane * 4];
  atomAddF(o + 0, v.x);
  atomAddF(o + 1, v.y);
  atomAddF(o + 2, v.z);
  atomAddF(o + 3, v.w);
}

// ---------------- WMMA fp32 GEMM: out = relu((A * rs_in[row]) @ W + b) [* post_scale[row]]
// One wave computes a 16x16 tile. K = 128 fixed. N templated (128 or 64).
template <int N>
__global__ void __launch_bounds__(256)
k_gemm(const float* __restrict__ A, const float* __restrict__ rs_in,
       const float* __restrict__ W, const float* __restrict__ bias,
       const float* __restrict__ post_scale, float* __restrict__ out, int M) {
  constexpr int K = 128;
  constexpr int ntiles = N / 16;
  const int lane = threadIdx.x & 31;
  const int wave = (int)(((long)blockIdx.x * blockDim.x + threadIdx.x) >> 5);
  const int mt = wave / ntiles;
  const int nt = wave % ntiles;
  if (mt * 16 >= M) return;
  const int m0 = mt * 16;
  const int n0 = nt * 16;

  // A 16x4 fp32 fragment: lane%16 = row M; lanes<16 hold K+{0,1}, lanes>=16 hold K+{2,3}
  const int row = m0 + (lane & 15);
  const int col = n0 + (lane & 15);
  const int kh  = (lane >> 4) * 2;
  const float s = rs_in[row];
  const float* __restrict__ arow = A + (size_t)row * K;

  v8f c = {};
#pragma unroll
  for (int kk = 0; kk < K; kk += 4) {
    v2f a;
    const v2f av = *(const v2f*)&arow[kk + kh];       // contiguous float2 load
    a.x = av.x * s;
    a.y = av.y * s;
    v2f b;
    b.x = W[(kk + kh) * N + col];                     // B 4x16: lane%16 = col N
    b.y = W[(kk + kh + 1) * N + col];
    c = __builtin_amdgcn_wmma_f32_16x16x4_f32(false, a, false, b, (short)0, c,
                                              false, false);
  }

  const float bz = bias[col];
#pragma unroll
  for (int r = 0; r < 8; ++r) {
    const int rr = m0 + r + 8 * (lane >> 4);          // C/D layout: VGPR r, lane half
    float v = c[r] + bz;
    v = v > 0.f ? v : 0.f;
    if (post_scale) v *= post_scale[rr];              // fuse next layer's D_src^{-1/2}
    out[(size_t)rr * N + col] = v;
  }
}

// ---------------- per-graph counts ----------------
__global__ void k_count(const int* __restrict__ gid, float* cnt, int Nn) {
  int n = blockIdx.x * blockDim.x + threadIdx.x;
  if (n < Nn) atomAddF(&cnt[gid[n]], 1.0f);
}

// ---------------- segmented sum pooling (graph_ids sorted) ----------------
// Block handles 64 nodes; thread = (feature f, node sub-stride). Register-accumulate
// per graph segment, flush on change -> few atomics per block.
__global__ void __launch_bounds__(256)
k_pool(const float* __restrict__ h, const int* __restrict__ gid,
       float* __restrict__ pooled, int Nn) {
  int n0 = blockIdx.x * 64;
  int f = threadIdx.x & 63;
  int sub = threadIdx.x >> 6;
  float acc = 0.f;
  int cur = -1;
  int nend = min(n0 + 64, Nn);
  for (int n = n0 + sub; n < nend; n += 4) {
    int g = gid[n];
    if (g != cur) {
      if (cur >= 0) atomAddF(&pooled[cur * OUTF + f], acc);
      acc = 0.f;
      cur = g;
    }
    acc += h[(size_t)n * OUTF + f];
  }
  if (cur >= 0) atomAddF(&pooled[cur * OUTF + f], acc);
}

// ---------------- head: mean -> FC+relu -> FC -> softmax ----------------
__global__ void k_head(const float* __restrict__ pooled, const float* __restrict__ cnt,
                       const float* __restrict__ fcW1, const float* __restrict__ fcb1,
                       const float* __restrict__ fcW2, const float* __restrict__ fcb2,
                       float* __restrict__ out) {
  int g = threadIdx.x;
  if (g >= NGRAPHS) return;
  float mean[OUTF];
  float ic = 1.0f / fmaxf(cnt[g], 1.0f);
  for (int k = 0; k < OUTF; ++k) mean[k] = pooled[g * OUTF + k] * ic;
  float z[HID];
  for (int j = 0; j < HID; ++j) {
    float a = fcb1[j];
    for (int k = 0; k < OUTF; ++k) a += mean[k] * fcW1[k * HID + j];
    z[j] = fmaxf(a, 0.f);
  }
  float o[NCLS];
  float mx = -1e30f;
  for (int c = 0; c < NCLS; ++c) {
    float a = fcb2[c];
    for (int j = 0; j < HID; ++j) a += z[j] * fcW2[j * NCLS + c];
    o[c] = a;
    mx = fmaxf(mx, a);
  }
  float sum = 0.f;
  for (int c = 0; c < NCLS; ++c) { o[c] = expf(o[c] - mx); sum += o[c]; }
  float is = 1.f / sum;
  for (int c = 0; c < NCLS; ++c) out[g * NCLS + c] = o[c] * is;
}

extern "C" void kernel_launch(void* const* d_in, const int* in_sizes, int n_in,
                              void* d_out, int out_size, void* d_ws, size_t ws_size,
                              hipStream_t stream) {
  const float* feats = (const float*)d_in[0];
  const int*   src   = (const int*)d_in[1];
  const int*   dst   = (const int*)d_in[2];
  const int*   gid   = (const int*)d_in[3];
  const float* W1    = (const float*)d_in[4];
  const float* b1    = (const float*)d_in[5];
  const float* W2    = (const float*)d_in[6];
  const float* b2    = (const float*)d_in[7];
  const float* fcW1  = (const float*)d_in[8];
  const float* fcb1  = (const float*)d_in[9];
  const float* fcW2  = (const float*)d_in[10];
  const float* fcb2  = (const float*)d_in[11];
  float* out = (float*)d_out;

  // workspace carve-up (all 16B-aligned sizes)
  char* w = (char*)d_ws;
  int*   degO   = (int*)w;        w += (size_t)NNODES * 4;
  int*   degI   = (int*)w;        w += (size_t)NNODES * 4;
  float* rsO    = (float*)w;      w += (size_t)NNODES * 4;
  float* rsI    = (float*)w;      w += (size_t)NNODES * 4;
  float* bufA   = (float*)w;      w += (size_t)NNODES * HF * 4;   // scaled feats / X1
  float* bufB   = (float*)w;      w += (size_t)NNODES * HF * 4;   // aggregation target
  float* bufC   = (float*)w;      w += (size_t)NNODES * OUTF * 4; // H2
  float* pooled = (float*)w;      w += (size_t)NGRAPHS * OUTF * 4;
  float* cnt    = (float*)w;      w += (size_t)NGRAPHS * 4;

  const int T = 256;

  // degrees
  k_zero4<<<((2L * NNODES / 4) + T - 1) / T, T, 0, stream>>>((float4*)degO, 2L * NNODES / 4);
  k_deg<<<(NEDGES + T - 1) / T, T, 0, stream>>>(src, dst, degO, degI, NEDGES);
  k_rsq<<<(NNODES + T - 1) / T, T, 0, stream>>>(degO, degI, rsO, rsI, NNODES);

  // layer 1: X0 = feats * rsO ; AGG = scatter(X0) ; X1 = relu(AGG*rsI @ W1 + b1) * rsO
  k_scale<<<((long)NNODES * (INF / 4) + T - 1) / T, T, 0, stream>>>(feats, rsO, bufA, NNODES);
  k_zero4<<<((long)NNODES * HF / 4 + T - 1) / T, T, 0, stream>>>((float4*)bufB, (long)NNODES * HF / 4);
  k_edge_agg<<<((long)NEDGES * 32 + T - 1) / T, T, 0, stream>>>(bufA, src, dst, bufB, NEDGES);
  {
    long waves = (long)(NNODES / 16) * (HF / 16);
    k_gemm<HF><<<(waves * 32 + T - 1) / T, T, 0, stream>>>(bufB, rsI, W1, b1, rsO, bufA, NNODES);
  }

  // layer 2: AGG = scatter(X1) ; H2 = relu(AGG*rsI @ W2 + b2)
  k_zero4<<<((long)NNODES * HF / 4 + T - 1) / T, T, 0, stream>>>((float4*)bufB, (long)NNODES * HF / 4);
  k_edge_agg<<<((long)NEDGES * 32 + T - 1) / T, T, 0, stream>>>(bufA, src, dst, bufB, NEDGES);
  {
    long waves = (long)(NNODES / 16) * (OUTF / 16);
    k_gemm<OUTF><<<(waves * 32 + T - 1) / T, T, 0, stream>>>(bufB, rsI, W2, b2, nullptr, bufC, NNODES);
  }

  // pooling + head
  long poolz4 = ((long)NGRAPHS * OUTF + NGRAPHS) / 4;   // pooled + cnt contiguous
  k_zero4<<<(poolz4 + T - 1) / T, T, 0, stream>>>((float4*)pooled, poolz4);
  k_count<<<(NNODES + T - 1) / T, T, 0, stream>>>(gid, cnt, NNODES);
  k_pool<<<(NNODES + 63) / 64, T, 0, stream>>>(bufC, gid, pooled, NNODES);
  k_head<<<1, NGRAPHS, 0, stream>>>(pooled, cnt, fcW1, fcb1, fcW2, fcb2, out);
}